// GCN_50379966382598
// MI455X (gfx1250) — compile-verified
//
#include <hip/hip_runtime.h>
#include <hip/hip_bf16.h>

typedef __attribute__((ext_vector_type(2))) float v2f;
typedef __attribute__((ext_vector_type(8))) float v8f;

#define CIN 128   // all three layers have 128 input channels
#define LDP 130   // LDS row stride (floats): bank (2c+k)%64 -> conflict-free b64 reads

// ---------------------------------------------------------------------------
// Degree / normalization
// ---------------------------------------------------------------------------
__global__ void deg_init(float* __restrict__ deg, int N) {
    int i = blockIdx.x * blockDim.x + threadIdx.x;
    if (i < N) deg[i] = 1.0f;               // self-loop contributes 1
}

__global__ void deg_accum(const int* __restrict__ col, float* __restrict__ deg, int E) {
    int e = blockIdx.x * blockDim.x + threadIdx.x;
    if (e < E) atomicAdd(&deg[col[e]], 1.0f);
}

__global__ void deg_rsqrt(float* __restrict__ dinv, int N) {
    int i = blockIdx.x * blockDim.x + threadIdx.x;
    if (i < N) dinv[i] = rsqrtf(dinv[i]);   // deg >= 1 always (self loop)
}

// ---------------------------------------------------------------------------
// Zero-pad W3 (128 x 40) to (128 x 48) so the GEMM needs no column guards.
// ---------------------------------------------------------------------------
__global__ void pad_w(const float* __restrict__ Wsrc, float* __restrict__ Wdst,
                      int Cin, int C, int CP) {
    int idx = blockIdx.x * blockDim.x + threadIdx.x;
    if (idx >= Cin * CP) return;
    int k = idx / CP, c = idx % CP;
    Wdst[idx] = (c < C) ? Wsrc[k * C + c] : 0.0f;
}

// ---------------------------------------------------------------------------
// GEMM: H[N x CP] = A[N x 128] * W[128 x CP], CP = COLTILES*16.
// Block = ROWTILES 16-row tiles (64 rows); blockDim = 32*COLTILES.
// A tiles + full transposed W staged in LDS (stride-130, conflict-free b64).
// Inner loop (guard-free, fully unrolled): one B fragment feeds ROWTILES
// independent WMMA accumulator chains -> no WMMA->WMMA RAW hazards, 4x B reuse,
// 4x less W-staging traffic than one-row-tile blocks.
// ---------------------------------------------------------------------------
template <int COLTILES, int ROWTILES>
__global__ void gcn_gemm_wmma(const float* __restrict__ A, const float* __restrict__ W,
                              float* __restrict__ H, int N) {
    constexpr int CP = COLTILES * 16;
    constexpr int MR = ROWTILES * 16;
    __shared__ float As[MR * LDP];
    __shared__ float Wt[CP * LDP];

    const int rowBase  = blockIdx.x * MR;
    const int tid      = threadIdx.x;
    const int nThreads = blockDim.x;

    // Stage MR x 128 A tiles (row-major, padded stride)
    for (int i = tid; i < MR * CIN; i += nThreads) {
        int r = i >> 7, k = i & (CIN - 1);
        int gr = rowBase + r;
        As[r * LDP + k] = (gr < N) ? A[(size_t)gr * CIN + k] : 0.0f;
    }
    // Stage transposed W: Wt[c][k] = W[k][c]; global reads coalesced over c
    for (int i = tid; i < CP * CIN; i += nThreads) {
        int c = i % CP, k = i / CP;
        Wt[c * LDP + k] = W[(size_t)k * CP + c];
    }
    __syncthreads();

    const int wave = tid >> 5;
    const int lane = tid & 31;
    const int half = lane >> 4;          // 0: lanes 0-15, 1: lanes 16-31
    const int mn   = lane & 15;          // M for A-frag, N for B-frag
    const float* aBase = &As[mn * LDP + 2 * half];
    const float* bBase = &Wt[(wave * 16 + mn) * LDP + 2 * half];

    v8f acc[ROWTILES];
#pragma unroll
    for (int rt = 0; rt < ROWTILES; ++rt) acc[rt] = (v8f){};

#pragma unroll
    for (int kk = 0; kk < CIN; kk += 4) {
        v2f b = *reinterpret_cast<const v2f*>(bBase + kk);                 // ds_load_b64
#pragma unroll
        for (int rt = 0; rt < ROWTILES; ++rt) {
            v2f a = *reinterpret_cast<const v2f*>(aBase + rt * 16 * LDP + kk);
            acc[rt] = __builtin_amdgcn_wmma_f32_16x16x4_f32(
                false, a, false, b, (short)0, acc[rt], false, false);
        }
    }

    // C/D layout: VGPR v -> M = v + 8*half, N = mn
    const int c = wave * 16 + mn;
#pragma unroll
    for (int rt = 0; rt < ROWTILES; ++rt) {
#pragma unroll
        for (int v = 0; v < 8; ++v) {
            int r = rowBase + rt * 16 + v + half * 8;
            if (r < N) H[(size_t)r * CP + c] = acc[rt][v];
        }
    }
}

// ---------------------------------------------------------------------------
// out[i,c] = bias[c] + H[i,c] * dinv[i]^2   (self-loop fused with bias init)
// H has row stride Hs (padded); out has row stride C.
// ---------------------------------------------------------------------------
__global__ void agg_init(const float* __restrict__ H, int Hs,
                         const float* __restrict__ dinv, const float* __restrict__ bias,
                         float* __restrict__ out, int N, int C) {
    long long idx = (long long)blockIdx.x * blockDim.x + threadIdx.x;
    if (idx >= (long long)N * C) return;
    int i  = (int)(idx / C);
    int cc = (int)(idx % C);
    float di = dinv[i];
    out[idx] = bias[cc] + H[(size_t)i * Hs + cc] * di * di;
}

// ---------------------------------------------------------------------------
// Edge scatter: one wave per edge. out[col] += H[row] * dinv[row]*dinv[col]
// float4 gathers (global_load_b128), scalar f32 atomics. Requires C % 4 == 0.
// ---------------------------------------------------------------------------
__global__ void agg_scatter(const float* __restrict__ H, int Hs,
                            const int* __restrict__ row, const int* __restrict__ col,
                            const float* __restrict__ dinv, float* __restrict__ out,
                            int E, int C) {
    int gid  = blockIdx.x * blockDim.x + threadIdx.x;
    int e    = gid >> 5;
    int lane = threadIdx.x & 31;
    if (e >= E) return;
    int r = row[e];
    int d = col[e];
    float nrm = dinv[r] * dinv[d];
    const float* hr = H + (size_t)r * Hs;
    float* od = out + (size_t)d * C;
    for (int c = lane * 4; c < C; c += 128) {
        float4 v = *reinterpret_cast<const float4*>(hr + c);
        atomicAdd(&od[c + 0], v.x * nrm);
        atomicAdd(&od[c + 1], v.y * nrm);
        atomicAdd(&od[c + 2], v.z * nrm);
        atomicAdd(&od[c + 3], v.w * nrm);
    }
}

__global__ void relu_inplace(float* __restrict__ x, long long n) {
    long long i = (long long)blockIdx.x * blockDim.x + threadIdx.x;
    if (i < n) x[i] = fmaxf(x[i], 0.0f);
}

// ---------------------------------------------------------------------------
// Row-wise log-softmax, one wave per row (wave32 shuffle reductions)
// ---------------------------------------------------------------------------
__global__ void log_softmax_rows(const float* __restrict__ in, float* __restrict__ out,
                                 int N, int C) {
    int gid  = blockIdx.x * blockDim.x + threadIdx.x;
    int node = gid >> 5;
    int lane = threadIdx.x & 31;
    if (node >= N) return;
    const float* xr = in + (size_t)node * C;
    float m = -3.402823466e38f;
    for (int c = lane; c < C; c += 32) m = fmaxf(m, xr[c]);
    for (int off = 16; off; off >>= 1) m = fmaxf(m, __shfl_xor(m, off, 32));
    float s = 0.0f;
    for (int c = lane; c < C; c += 32) s += expf(xr[c] - m);
    for (int off = 16; off; off >>= 1) s += __shfl_xor(s, off, 32);
    float lse = m + logf(s);
    float* orow = out + (size_t)node * C;
    for (int c = lane; c < C; c += 32) orow[c] = xr[c] - lse;
}

// ---------------------------------------------------------------------------
extern "C" void kernel_launch(void* const* d_in, const int* in_sizes, int n_in,
                              void* d_out, int out_size, void* d_ws, size_t ws_size,
                              hipStream_t stream) {
    const float* x   = (const float*)d_in[0];
    const int*   ei  = (const int*)d_in[1];
    const float* W1  = (const float*)d_in[2];
    const float* b1  = (const float*)d_in[3];
    const float* W2  = (const float*)d_in[4];
    const float* b2  = (const float*)d_in[5];
    const float* W3  = (const float*)d_in[6];
    const float* b3  = (const float*)d_in[7];

    const int N = in_sizes[0] / CIN;       // 100000
    const int E = in_sizes[1] / 2;         // 1600000
    const int* row = ei;                   // edge_index[0] = source
    const int* col = ei + E;               // edge_index[1] = target

    // Workspace (floats): dinv[N] | bufH[N*128] | bufA[N*128] | Wpad[128*48]
    float* dinv = (float*)d_ws;
    float* bufH = dinv + (((size_t)N + 255) & ~(size_t)255);
    float* bufA = bufH + (size_t)N * CIN;
    float* Wpad = bufA + (size_t)N * CIN;

    const int TB = 256;
    const int gemmBlocks = (N + 63) / 64;          // 4 row tiles per block
    const long long nc128 = (long long)N * 128;
    const long long nc40  = (long long)N * 40;

    // --- normalization: deg -> dinv ---
    deg_init  <<<(N + TB - 1) / TB, TB, 0, stream>>>(dinv, N);
    deg_accum <<<(E + TB - 1) / TB, TB, 0, stream>>>(col, dinv, E);
    deg_rsqrt <<<(N + TB - 1) / TB, TB, 0, stream>>>(dinv, N);

    // --- pad W3 (128x40 -> 128x48) ---
    pad_w<<<(128 * 48 + TB - 1) / TB, TB, 0, stream>>>(W3, Wpad, 128, 40, 48);

    // --- layer 1: 128 -> 128, ReLU ---
    gcn_gemm_wmma<8, 4><<<gemmBlocks, 256, 0, stream>>>(x, W1, bufH, N);
    agg_init   <<<(nc128 + TB - 1) / TB, TB, 0, stream>>>(bufH, 128, dinv, b1, bufA, N, 128);
    agg_scatter<<<(E + 7) / 8, TB, 0, stream>>>(bufH, 128, row, col, dinv, bufA, E, 128);
    relu_inplace<<<(nc128 + TB - 1) / TB, TB, 0, stream>>>(bufA, nc128);

    // --- layer 2: 128 -> 128, ReLU ---
    gcn_gemm_wmma<8, 4><<<gemmBlocks, 256, 0, stream>>>(bufA, W2, bufH, N);
    agg_init   <<<(nc128 + TB - 1) / TB, TB, 0, stream>>>(bufH, 128, dinv, b2, bufA, N, 128);
    agg_scatter<<<(E + 7) / 8, TB, 0, stream>>>(bufH, 128, row, col, dinv, bufA, E, 128);
    relu_inplace<<<(nc128 + TB - 1) / TB, TB, 0, stream>>>(bufA, nc128);

    // --- layer 3: 128 -> 40 (padded to 48), log-softmax ---
    gcn_gemm_wmma<3, 4><<<gemmBlocks, 96, 0, stream>>>(bufA, Wpad, bufH, N);
    agg_init   <<<(nc40 + TB - 1) / TB, TB, 0, stream>>>(bufH, 48, dinv, b3, bufA, N, 40);
    agg_scatter<<<(E + 7) / 8, TB, 0, stream>>>(bufH, 48, row, col, dinv, bufA, E, 40);
    log_softmax_rows<<<(N + 7) / 8, TB, 0, stream>>>(bufA, (float*)d_out, N, 40);
}